// MuonMambaVisionMixer_57964878627190
// MI455X (gfx1250) — compile-verified
//
#include <hip/hip_runtime.h>
#include <hip/hip_bf16.h>
#include <math.h>

#define D_MODEL 256
#define D_STATE 16
#define D_INNER 512
#define D_SSM   256
#define BATCH   8
#define SEQLEN  4096
#define M_TOTAL (BATCH*SEQLEN)   // 32768 rows

#define NS_Ac 3.4445f
#define NS_Bc (-4.7750f)
#define NS_Cc 2.0315f
#define BETA  0.9f
#define ALPHA 1.0f
#define HSTR  17                 // padded LDS row stride for h (bank-conflict free)

typedef float v2f __attribute__((ext_vector_type(2)));
typedef float v8f __attribute__((ext_vector_type(8)));

__device__ __forceinline__ float silu_f(float x)    { return x / (1.0f + expf(-x)); }
__device__ __forceinline__ float softplus_f(float x){ return (x > 20.0f) ? x : log1pf(expf(x)); }

__device__ __forceinline__ v8f v8f_zero() {
  v8f z;
  #pragma unroll
  for (int r = 0; r < 8; ++r) z[r] = 0.f;
  return z;
}

// 16x16 f32 output tile, K-loop of V_WMMA_F32_16X16X4_F32.
// A: row-major [16 x K] tile base (lda = row stride)
// W: row-major [16 x K] tile base, W[n][k] supplies B[k][n] (ldw = row stride)
__device__ __forceinline__ v8f wmma_gemm_acc(const float* __restrict__ Atile, int lda,
                                             const float* __restrict__ Wtile, int ldw,
                                             int K, v8f acc) {
  const int lane = threadIdx.x & 31;
  const int r  = lane & 15;          // M row (A) / N col (B)
  const int kb = (lane >> 4) << 1;   // lanes 16-31 hold k+2,k+3
  const float* __restrict__ ap = Atile + (size_t)r * lda + kb;
  const float* __restrict__ wp = Wtile + (size_t)r * ldw + kb;
  #pragma unroll 8
  for (int k = 0; k < K; k += 4) {
    v2f a; a.x = ap[k]; a.y = ap[k + 1];
    v2f b; b.x = wp[k]; b.y = wp[k + 1];
    acc = __builtin_amdgcn_wmma_f32_16x16x4_f32(false, a, false, b, (short)0, acc, false, false);
  }
  return acc;
}

__device__ __forceinline__ void store_tile(float* __restrict__ Cp, int ldc, v8f c) {
  const int lane = threadIdx.x & 31;
  const int n  = lane & 15;
  const int mh = (lane >> 4) * 8;    // VGPR r: M = r (lanes 0-15) / r+8 (lanes 16-31)
  #pragma unroll
  for (int r = 0; r < 8; ++r) Cp[(size_t)(mh + r) * ldc + n] = c[r];
}

// ---------------- K1: in_proj  xz[m, 0..511] = H[m,:] @ Win^T ----------------
__global__ __launch_bounds__(128)
void k_inproj(const float* __restrict__ H, const float* __restrict__ W,
              float* __restrict__ out) {
  int wave = (blockIdx.x * blockDim.x + threadIdx.x) >> 5;
  int nt = wave & 31;                // 512/16 N tiles
  int mt = wave >> 5;                // 2048 M tiles
  v8f c = v8f_zero();
  c = wmma_gemm_acc(H + (size_t)mt * 16 * D_MODEL, D_MODEL,
                    W + (size_t)nt * 16 * D_MODEL, D_MODEL, D_MODEL, c);
  store_tile(out + (size_t)mt * 16 * D_INNER + nt * 16, D_INNER, c);
}

// ---------------- K2: depthwise conv (pad 1,2) + SiLU ----------------
__global__ __launch_bounds__(256)
void k_conv(const float* __restrict__ xz,
            const float* __restrict__ wx, const float* __restrict__ bx,
            const float* __restrict__ wz, const float* __restrict__ bz,
            float* __restrict__ xo, float* __restrict__ zo) {
  int idx = blockIdx.x * blockDim.x + threadIdx.x;     // B*L*256
  int d = idx & 255;
  int m = idx >> 8;                                    // b*L + l
  int l = m & (SEQLEN - 1);
  float ax = bx[d], az = bz[d];
  #pragma unroll
  for (int j = 0; j < 4; ++j) {
    int ll = l - 1 + j;
    if (ll >= 0 && ll < SEQLEN) {
      const float* row = xz + (size_t)(m - l + ll) * D_INNER;
      ax += wx[d * 4 + j] * row[d];
      az += wz[d * 4 + j] * row[D_SSM + d];
    }
  }
  xo[idx] = silu_f(ax);
  zo[idx] = silu_f(az);
}

// -------- K2b: combined weight Wc[d][k] = sum_r dtw[d][r] * xpw[r][k] --------
__global__ __launch_bounds__(256)
void k_wc(const float* __restrict__ dtw, const float* __restrict__ xpw,
          float* __restrict__ Wc) {
  int idx = blockIdx.x * blockDim.x + threadIdx.x;     // 256*256
  int dch = idx >> 8, n = idx & 255;
  float acc = 0.f;
  #pragma unroll
  for (int r = 0; r < 16; ++r) acc += dtw[dch * 16 + r] * xpw[r * 256 + n];
  Wc[idx] = acc;
}

// -------- K3: delta = softplus(x @ Wc^T + dt_bias) --------
__global__ __launch_bounds__(128)
void k_delta(const float* __restrict__ X, const float* __restrict__ Wc,
             const float* __restrict__ dtb, float* __restrict__ delta) {
  int wave = (blockIdx.x * blockDim.x + threadIdx.x) >> 5;
  int nt = wave & 15, mt = wave >> 4;
  v8f c = v8f_zero();
  c = wmma_gemm_acc(X + (size_t)mt * 16 * D_SSM, D_SSM,
                    Wc + (size_t)nt * 16 * D_SSM, D_SSM, D_SSM, c);
  const int lane = threadIdx.x & 31;
  const int n = lane & 15, mh = (lane >> 4) * 8;
  float bias = dtb[nt * 16 + n];
  float* out = delta + (size_t)mt * 16 * D_SSM + nt * 16;
  #pragma unroll
  for (int r = 0; r < 8; ++r) out[(size_t)(mh + r) * D_SSM + n] = softplus_f(c[r] + bias);
}

// -------- K3b: B,C = x @ x_proj_w[16:48]^T --------
__global__ __launch_bounds__(128)
void k_bc(const float* __restrict__ X, const float* __restrict__ xpw,
          float* __restrict__ Bc, float* __restrict__ Cc) {
  int wave = (blockIdx.x * blockDim.x + threadIdx.x) >> 5;
  int nt = wave & 1, mt = wave >> 1;
  v8f c = v8f_zero();
  c = wmma_gemm_acc(X + (size_t)mt * 16 * D_SSM, D_SSM,
                    xpw + (size_t)(16 + nt * 16) * D_SSM, D_SSM, D_SSM, c);
  float* out = (nt == 0 ? Bc : Cc) + (size_t)mt * 16 * 16;
  store_tile(out, 16, c);
}

// -------- K4: sequential selective scan + Newton-Schulz (WMMA inside) --------
__global__ __launch_bounds__(256)
void k_scan(const float* __restrict__ x, const float* __restrict__ delta,
            const float* __restrict__ Bcg, const float* __restrict__ Ccg,
            const float* __restrict__ A_log, const float* __restrict__ Dv,
            float* __restrict__ y) {
  const int b = blockIdx.x;
  const int tid = threadIdx.x;
  const int lane = tid & 31;
  const int w = tid >> 5;

  __shared__ float h[D_SSM * HSTR];    // state, row-padded
  __shared__ float Sp[8 * 256];        // per-wave Gram partials
  __shared__ float Sf[256];
  __shared__ float A16[256];
  __shared__ float M16[256];
  __shared__ float Bt[16], Ct[16];
  __shared__ float scal[2];            // 1/nrm, 1/nrm^2

  float Arow[16];
  #pragma unroll
  for (int n = 0; n < 16; ++n) Arow[n] = -expf(A_log[tid * 16 + n]);
  float vmom[16];
  #pragma unroll
  for (int n = 0; n < 16; ++n) { vmom[n] = 0.f; h[tid * HSTR + n] = 0.f; }
  const float Dd = Dv[tid];

  const float* xb = x     + (size_t)b * SEQLEN * D_SSM;
  const float* db = delta + (size_t)b * SEQLEN * D_SSM;
  const float* Bb = Bcg   + (size_t)b * SEQLEN * 16;
  const float* Cb = Ccg   + (size_t)b * SEQLEN * 16;
  float*       yb = y     + (size_t)b * SEQLEN * D_SSM;

  // software prefetch of t=0
  float xt = xb[tid], dtt = db[tid];
  float bcreg = 0.f;
  if (tid < 16) bcreg = Bb[tid];
  else if (tid < 32) bcreg = Cb[tid - 16];
  __syncthreads();

  for (int t = 0; t < SEQLEN; ++t) {
    if (tid < 16) Bt[tid] = bcreg;
    else if (tid < 32) Ct[tid - 16] = bcreg;
    float xcur = xt, dcur = dtt;
    if (t + 1 < SEQLEN) {                       // prefetch next step
      xt  = xb[(size_t)(t + 1) * D_SSM + tid];
      dtt = db[(size_t)(t + 1) * D_SSM + tid];
      if (tid < 16) bcreg = Bb[(size_t)(t + 1) * 16 + tid];
      else if (tid < 32) bcreg = Cb[(size_t)(t + 1) * 16 + tid - 16];
    }
    __syncthreads();                            // Bt/Ct ready

    // state update: h[d] = exp(dt*A)*h + (0.9 v + dt*x*B)
    {
      float dx = dcur * xcur;
      #pragma unroll
      for (int n = 0; n < 16; ++n) {
        float dA = expf(dcur * Arow[n]);
        vmom[n] = BETA * vmom[n] + ALPHA * dx * Bt[n];
        h[tid * HSTR + n] = dA * h[tid * HSTR + n] + vmom[n];
      }
    }
    __syncthreads();

    // Gram S = X X^T (16x16), K=256 split across 8 waves; A==B per lane
    {
      v8f s = v8f_zero();
      const int i  = lane & 15;
      const int kb = (lane >> 4) << 1;
      const int d0 = w * 32;
      #pragma unroll
      for (int kk = 0; kk < 8; ++kk) {
        int dd = d0 + kk * 4 + kb;
        v2f a; a.x = h[dd * HSTR + i]; a.y = h[(dd + 1) * HSTR + i];
        s = __builtin_amdgcn_wmma_f32_16x16x4_f32(false, a, false, a, (short)0, s, false, false);
      }
      const int n = lane & 15, mh = (lane >> 4) * 8;
      #pragma unroll
      for (int r = 0; r < 8; ++r) Sp[w * 256 + (mh + r) * 16 + n] = s[r];
    }
    __syncthreads();

    { // reduce 8 partials
      float acc = 0.f;
      #pragma unroll
      for (int ww = 0; ww < 8; ++ww) acc += Sp[ww * 256 + tid];
      Sf[tid] = acc;
    }
    __syncthreads();
    if (tid == 0) {                              // ||h||_F = sqrt(trace(S))
      float tr = 0.f;
      #pragma unroll
      for (int i2 = 0; i2 < 16; ++i2) tr += Sf[i2 * 17];
      float nrm = sqrtf(tr) + 1e-7f;
      scal[0] = 1.0f / nrm;
      scal[1] = 1.0f / (nrm * nrm);
    }
    __syncthreads();
    A16[tid] = Sf[tid] * scal[1];                // A = Xn Xn^T
    __syncthreads();
    {                                            // M16 = (b*A + c*A^2) * (1/nrm)
      const int i = tid >> 4, j = tid & 15;
      float acc = 0.f;
      #pragma unroll
      for (int k = 0; k < 16; ++k) acc += A16[i * 16 + k] * A16[k * 16 + j];
      M16[tid] = (NS_Bc * A16[tid] + NS_Cc * acc) * scal[0];
    }
    __syncthreads();

    // apply: Xnew = a*Xn + M16 @ h  (1/nrm folded into M16 and a-scale)
    {
      const float ascale = NS_Ac * scal[0];
      const int i  = lane & 15;
      const int kb = (lane >> 4) << 1;
      const int n = lane & 15, mh = (lane >> 4) * 8;
      #pragma unroll
      for (int half = 0; half < 2; ++half) {
        const int d0 = w * 32 + half * 16;
        v8f c;
        #pragma unroll
        for (int r = 0; r < 8; ++r) c[r] = ascale * h[(d0 + n) * HSTR + mh + r];
        #pragma unroll
        for (int k0 = 0; k0 < 16; k0 += 4) {
          v2f a;  a.x = M16[i * 16 + k0 + kb];         a.y = M16[i * 16 + k0 + kb + 1];
          v2f bb; bb.x = h[(d0 + n) * HSTR + k0 + kb]; bb.y = h[(d0 + n) * HSTR + k0 + kb + 1];
          c = __builtin_amdgcn_wmma_f32_16x16x4_f32(false, a, false, bb, (short)0, c, false, false);
        }
        #pragma unroll
        for (int r = 0; r < 8; ++r) h[(d0 + n) * HSTR + mh + r] = c[r];
      }
    }
    __syncthreads();

    { // y = h . C + D*x
      float acc = Dd * xcur;
      #pragma unroll
      for (int n = 0; n < 16; ++n) acc += h[tid * HSTR + n] * Ct[n];
      yb[(size_t)t * D_SSM + tid] = acc;
    }
    __syncthreads();
  }
}

// -------- K5: out = [y | z] @ Wout^T (two K-passes, no concat buffer) --------
__global__ __launch_bounds__(128)
void k_out(const float* __restrict__ Y, const float* __restrict__ Z,
           const float* __restrict__ Wout, float* __restrict__ out) {
  int wave = (blockIdx.x * blockDim.x + threadIdx.x) >> 5;
  int nt = wave & 15, mt = wave >> 4;
  v8f c = v8f_zero();
  c = wmma_gemm_acc(Y + (size_t)mt * 16 * D_SSM, D_SSM,
                    Wout + (size_t)nt * 16 * D_INNER, D_INNER, D_SSM, c);
  c = wmma_gemm_acc(Z + (size_t)mt * 16 * D_SSM, D_SSM,
                    Wout + (size_t)nt * 16 * D_INNER + D_SSM, D_INNER, D_SSM, c);
  store_tile(out + (size_t)mt * 16 * D_MODEL + nt * 16, D_MODEL, c);
}

extern "C" void kernel_launch(void* const* d_in, const int* in_sizes, int n_in,
                              void* d_out, int out_size, void* d_ws, size_t ws_size,
                              hipStream_t stream) {
  const float* hidden     = (const float*)d_in[0];
  const float* in_proj_w  = (const float*)d_in[1];
  const float* conv_x_w   = (const float*)d_in[2];
  const float* conv_x_b   = (const float*)d_in[3];
  const float* conv_z_w   = (const float*)d_in[4];
  const float* conv_z_b   = (const float*)d_in[5];
  const float* x_proj_w   = (const float*)d_in[6];
  const float* dt_proj_w  = (const float*)d_in[7];
  const float* dt_proj_b  = (const float*)d_in[8];
  const float* A_log      = (const float*)d_in[9];
  const float* Dv         = (const float*)d_in[10];
  const float* out_proj_w = (const float*)d_in[11];

  float* ws   = (float*)d_ws;
  float* xz   = ws;                  // 16,777,216 f (also reused as y after conv)
  float* xbuf = ws + 16777216;       //  8,388,608 f
  float* zbuf = ws + 25165824;       //  8,388,608 f
  float* dbuf = ws + 33554432;       //  8,388,608 f
  float* Bc   = ws + 41943040;       //    524,288 f
  float* Cc   = ws + 42467328;       //    524,288 f
  float* Wc   = ws + 42991616;       //     65,536 f
  float* ybuf = xz;                  // xz is dead after conv

  k_inproj<<<(2048 * 32) / 4, 128, 0, stream>>>(hidden, in_proj_w, xz);
  k_conv<<<(M_TOTAL * D_SSM) / 256, 256, 0, stream>>>(xz, conv_x_w, conv_x_b,
                                                      conv_z_w, conv_z_b, xbuf, zbuf);
  k_wc<<<256, 256, 0, stream>>>(dt_proj_w, x_proj_w, Wc);
  k_delta<<<(2048 * 16) / 4, 128, 0, stream>>>(xbuf, Wc, dt_proj_b, dbuf);
  k_bc<<<(2048 * 2) / 4, 128, 0, stream>>>(xbuf, x_proj_w, Bc, Cc);
  k_scan<<<BATCH, 256, 0, stream>>>(xbuf, dbuf, Bc, Cc, A_log, Dv, ybuf);
  k_out<<<(2048 * 16) / 4, 128, 0, stream>>>(ybuf, zbuf, out_proj_w, (float*)d_out);
}